// TracesEncoder_11287174054679
// MI455X (gfx1250) — compile-verified
//
#include <hip/hip_runtime.h>

#define N_NODES 50000
#define N_EDGES 800000
#define HID_DIM 64
#define OUT_DIM 32
#define N_GRAPHS 512

typedef float v2f __attribute__((ext_vector_type(2)));
typedef float v8f __attribute__((ext_vector_type(8)));

// ---------------- utility ----------------
__global__ void gcn_zero_f32(float* __restrict__ p, int n) {
    int i = blockIdx.x * blockDim.x + threadIdx.x;
    if (i < n) p[i] = 0.0f;
}

// ---------------- WMMA GEMM: H[M x N] = X[M x K] @ W[K x N] (+ optional bias) ----------------
// One wave32 computes a 16-row x N-col tile using V_WMMA_F32_16X16X4_F32.
// A frag (16x4 f32): lane l holds row l%16, K = 2*(l/16)+v  -> contiguous float2 per lane.
// B frag (4x16 f32): b[v] = W[(k0+2*half+v)*N + n], n = nt*16 + l%16.
// C/D (16x16 f32):   vgpr r -> row r + 8*(l/16), col l%16.
template <int K, int N>
__global__ __launch_bounds__(256) void gcn_gemm_wmma(const float* __restrict__ X,
                                                     const float* __restrict__ W,
                                                     const float* __restrict__ bias,
                                                     float* __restrict__ H, int mtiles) {
    int wid = (blockIdx.x * blockDim.x + threadIdx.x) >> 5;   // wave id
    if (wid >= mtiles) return;                                 // whole wave exits -> EXEC all 1s
    int lane = threadIdx.x & 31;
    int half = lane >> 4;      // 0 or 1
    int lr   = lane & 15;      // 0..15

    constexpr int NT = N / 16;
    v8f zero = {};
    v8f acc[NT];
#pragma unroll
    for (int t = 0; t < NT; ++t) acc[t] = zero;

    int m0 = wid * 16;
    const float* xrow = X + (size_t)(m0 + lr) * K;

#pragma unroll
    for (int k0 = 0; k0 < K; k0 += 4) {
        int kk = k0 + 2 * half;
        v2f a;
        a.x = xrow[kk];
        a.y = xrow[kk + 1];
#pragma unroll
        for (int t = 0; t < NT; ++t) {
            int n = t * 16 + lr;
            v2f b;
            b.x = W[(size_t)kk * N + n];
            b.y = W[(size_t)(kk + 1) * N + n];
            acc[t] = __builtin_amdgcn_wmma_f32_16x16x4_f32(
                false, a, false, b, (short)0, acc[t], false, false);
        }
    }

#pragma unroll
    for (int t = 0; t < NT; ++t) {
        int n = t * 16 + lr;
        float bv = bias ? bias[n] : 0.0f;
#pragma unroll
        for (int r = 0; r < 8; ++r) {
            H[(size_t)(m0 + r + 8 * half) * N + n] = acc[t][r] + bv;
        }
    }
}

// ---------------- degree / normalization ----------------
__global__ void gcn_deg_count(const int* __restrict__ dst, float* __restrict__ deg) {
    int e = blockIdx.x * blockDim.x + threadIdx.x;
    if (e < N_EDGES) unsafeAtomicAdd(&deg[dst[e]], 1.0f);
}

__global__ void gcn_inv_sqrt(const float* __restrict__ deg, float* __restrict__ isq) {
    int i = blockIdx.x * blockDim.x + threadIdx.x;
    if (i < N_NODES) isq[i] = rsqrtf(deg[i] + 1.0f);   // +1 self-loop
}

// ---------------- edge scatter: agg[dst] += h[src] * isq[src]*isq[dst] ----------------
// 16 threads per edge, float4 per thread (64 features).
__global__ __launch_bounds__(256) void gcn_edge_agg(const int* __restrict__ src,
                                                    const int* __restrict__ dst,
                                                    const float* __restrict__ isq,
                                                    const float* __restrict__ h,
                                                    float* __restrict__ agg) {
    int t = blockIdx.x * blockDim.x + threadIdx.x;
    if (t >= N_EDGES * 16) return;
    int e  = t >> 4;
    int fg = t & 15;
    int s = src[e], d = dst[e];
    float norm = isq[s] * isq[d];
    const float4* h4 = (const float4*)h;
    float4 v = h4[(size_t)s * 16 + fg];
    float* o = agg + (size_t)d * HID_DIM + fg * 4;
    unsafeAtomicAdd(o + 0, v.x * norm);
    unsafeAtomicAdd(o + 1, v.y * norm);
    unsafeAtomicAdd(o + 2, v.z * norm);
    unsafeAtomicAdd(o + 3, v.w * norm);
}

// ---------------- self-loop + bias + relu (in place on agg) ----------------
__global__ void gcn_selfloop_bias_relu(float* __restrict__ agg, const float* __restrict__ h,
                                       const float* __restrict__ isq,
                                       const float* __restrict__ bias) {
    int t = blockIdx.x * blockDim.x + threadIdx.x;
    if (t >= N_NODES * HID_DIM) return;
    int i = t >> 6, f = t & 63;
    float is = isq[i];
    float v = agg[t] + h[t] * (is * is) + bias[f];
    agg[t] = v > 0.0f ? v : 0.0f;
}

// ---------------- global mean pool ----------------
__global__ void gcn_pool_count(const int* __restrict__ batch, float* __restrict__ counts) {
    int i = blockIdx.x * blockDim.x + threadIdx.x;
    if (i < N_NODES) unsafeAtomicAdd(&counts[batch[i]], 1.0f);
}

__global__ void gcn_pool_sum(const int* __restrict__ batch, const float* __restrict__ h,
                             float* __restrict__ sums) {
    int t = blockIdx.x * blockDim.x + threadIdx.x;
    if (t >= N_NODES * HID_DIM) return;
    int i = t >> 6, f = t & 63;
    unsafeAtomicAdd(&sums[(size_t)batch[i] * HID_DIM + f], h[t]);
}

__global__ void gcn_pool_div(float* __restrict__ sums, const float* __restrict__ counts) {
    int t = blockIdx.x * blockDim.x + threadIdx.x;
    if (t >= N_GRAPHS * HID_DIM) return;
    int g = t >> 6;
    sums[t] /= fmaxf(counts[g], 1.0f);
}

static inline int cdiv(int a, int b) { return (a + b - 1) / b; }

extern "C" void kernel_launch(void* const* d_in, const int* in_sizes, int n_in,
                              void* d_out, int out_size, void* d_ws, size_t ws_size,
                              hipStream_t stream) {
    const float* x   = (const float*)d_in[0];   // 50000 x 64
    const float* W1  = (const float*)d_in[1];   // 64 x 64
    const float* b1  = (const float*)d_in[2];   // 64
    const float* W2  = (const float*)d_in[3];   // 64 x 64
    const float* b2  = (const float*)d_in[4];   // 64
    const float* fcW = (const float*)d_in[5];   // 64 x 32
    const float* fcb = (const float*)d_in[6];   // 32
    const int*   ei  = (const int*)d_in[7];     // 2 x 800000
    const int*   bat = (const int*)d_in[8];     // 50000
    const int* srcI = ei;
    const int* dstI = ei + N_EDGES;

    // workspace layout (floats)
    float* ws     = (float*)d_ws;
    float* A      = ws;                               // 50000*64  (linear-transform buffer)
    float* B      = A + (size_t)N_NODES * HID_DIM;    // 50000*64  (agg / activation buffer)
    float* deg    = B + (size_t)N_NODES * HID_DIM;    // 50000
    float* isq    = deg + N_NODES;                    // 50000
    float* pooled = isq + N_NODES;                    // 512*64
    float* counts = pooled + (size_t)N_GRAPHS * HID_DIM; // 512

    const int TPB = 256;
    const int mtilesN = N_NODES / 16;                 // 3125 (exact)
    const int gemmBlocks = cdiv(mtilesN * 32, TPB);
    const int nodeFeat = N_NODES * HID_DIM;           // 3.2M
    const int edgeThreads = N_EDGES * 16;             // 12.8M

    // ---- degree + normalization (shared by both layers) ----
    gcn_zero_f32<<<cdiv(N_NODES, TPB), TPB, 0, stream>>>(deg, N_NODES);
    gcn_deg_count<<<cdiv(N_EDGES, TPB), TPB, 0, stream>>>(dstI, deg);
    gcn_inv_sqrt<<<cdiv(N_NODES, TPB), TPB, 0, stream>>>(deg, isq);

    // ---- layer 1: A = x @ W1 ; B = scatter(A) ; B = relu(B + A/deg + b1) ----
    gcn_gemm_wmma<64, 64><<<gemmBlocks, TPB, 0, stream>>>(x, W1, nullptr, A, mtilesN);
    gcn_zero_f32<<<cdiv(nodeFeat, TPB), TPB, 0, stream>>>(B, nodeFeat);
    gcn_edge_agg<<<cdiv(edgeThreads, TPB), TPB, 0, stream>>>(srcI, dstI, isq, A, B);
    gcn_selfloop_bias_relu<<<cdiv(nodeFeat, TPB), TPB, 0, stream>>>(B, A, isq, b1);

    // ---- layer 2: A = B @ W2 ; B = scatter(A) ; B = relu(B + A/deg + b2) ----
    gcn_gemm_wmma<64, 64><<<gemmBlocks, TPB, 0, stream>>>(B, W2, nullptr, A, mtilesN);
    gcn_zero_f32<<<cdiv(nodeFeat, TPB), TPB, 0, stream>>>(B, nodeFeat);
    gcn_edge_agg<<<cdiv(edgeThreads, TPB), TPB, 0, stream>>>(srcI, dstI, isq, A, B);
    gcn_selfloop_bias_relu<<<cdiv(nodeFeat, TPB), TPB, 0, stream>>>(B, A, isq, b2);

    // ---- global mean pool ----
    gcn_zero_f32<<<cdiv(N_GRAPHS * HID_DIM + N_GRAPHS, TPB), TPB, 0, stream>>>(
        pooled, N_GRAPHS * HID_DIM + N_GRAPHS);  // pooled and counts are contiguous
    gcn_pool_count<<<cdiv(N_NODES, TPB), TPB, 0, stream>>>(bat, counts);
    gcn_pool_sum<<<cdiv(nodeFeat, TPB), TPB, 0, stream>>>(bat, B, pooled);
    gcn_pool_div<<<cdiv(N_GRAPHS * HID_DIM, TPB), TPB, 0, stream>>>(pooled, counts);

    // ---- FC: out = pooled @ fcW + fcb  (512x64 @ 64x32) ----
    const int mtilesG = N_GRAPHS / 16;  // 32
    gcn_gemm_wmma<64, 32><<<cdiv(mtilesG * 32, TPB), TPB, 0, stream>>>(
        pooled, fcW, fcb, (float*)d_out, mtilesG);
}